// ComputeLossOTA_21466246545765
// MI455X (gfx1250) — compile-verified
//
#include <hip/hip_runtime.h>

// ---------------------------------------------------------------------------
// YOLO ComputeLossOTA-style loss for MI455X (gfx1250), wave32.
//
// d_out = [loss*B, loss_box, loss_obj, loss_cls, loss]  (5 floats)
//
// ws layout (floats):
//   [0, 403200)           scat: scattered t_obj over all cells (3 levels)
//   [403200, 403296)      box partials  (3 levels x 32 blocks)
//   [403296, 403392)      cls partials  (3 levels x 32 blocks)
//   [403392, 407232)      obj partials  (3 levels x 1280 blocks)
// ---------------------------------------------------------------------------

typedef __attribute__((ext_vector_type(2))) float v2f;
typedef __attribute__((ext_vector_type(8))) float v8f;

#define NCLS 80
#define CH   85   // 5 + NC
#define EPSF 1e-7f

// ---------------------------------------------------------------------------
// Full-precision wave32 sum using V_WMMA_F32_16X16X4_F32.
// A (16x4 f32): lanes 0-15 hold (M=lane, K=0) in VGPR0, K=1 in VGPR1;
// lanes 16-31 hold (M=lane-16, K=2/3). We place each lane's value at its
// (M,K) slot with zeros elsewhere; B = ones(4x16) so D[m,n] = rowsum(m).
// D layout: VGPR r -> (M=r, N=lane) for lanes 0-15, (M=8+r) for lanes 16-31.
// Sum of d[0..7] gives half the rows; xor-16 shuffle adds the other half.
// ---------------------------------------------------------------------------
__device__ __forceinline__ float wmma_wave_sum(float x) {
  v2f a; a[0] = x;   a[1] = 0.0f;
  v2f b; b[0] = 1.0f; b[1] = 1.0f;
  v8f c = {};
  v8f d = __builtin_amdgcn_wmma_f32_16x16x4_f32(
      /*neg_a=*/false, a, /*neg_b=*/false, b,
      /*c_mod=*/(short)0, c, /*reuse_a=*/false, /*reuse_b=*/false);
  float s = d[0] + d[1] + d[2] + d[3] + d[4] + d[5] + d[6] + d[7];
  s += __shfl_xor(s, 16, 32);
  return s;  // full 32-lane sum in every lane
}

__device__ __forceinline__ float softplusf(float x) {
  // numerically stable: max(x,0) + log1p(exp(-|x|))
  return fmaxf(x, 0.0f) + log1pf(expf(-fabsf(x)));
}

__device__ __forceinline__ float sigmoidf(float x) {
  return 1.0f / (1.0f + expf(-x));
}

__device__ __forceinline__ float ciou_f(float px, float py, float pw, float ph,
                                        float tx, float ty, float tw, float th) {
  float b1x1 = px - pw * 0.5f, b1x2 = px + pw * 0.5f;
  float b1y1 = py - ph * 0.5f, b1y2 = py + ph * 0.5f;
  float b2x1 = tx - tw * 0.5f, b2x2 = tx + tw * 0.5f;
  float b2y1 = ty - th * 0.5f, b2y2 = ty + th * 0.5f;
  float iw = fmaxf(fminf(b1x2, b2x2) - fmaxf(b1x1, b2x1), 0.0f);
  float ih = fmaxf(fminf(b1y2, b2y2) - fmaxf(b1y1, b2y1), 0.0f);
  float inter = iw * ih;
  float uni   = pw * ph + tw * th - inter + EPSF;
  float iou   = inter / uni;
  float cw = fmaxf(b1x2, b2x2) - fminf(b1x1, b2x1);
  float ch = fmaxf(b1y2, b2y2) - fminf(b1y1, b2y1);
  float c2 = cw * cw + ch * ch + EPSF;
  float dx = b2x1 + b2x2 - b1x1 - b1x2;
  float dy = b2y1 + b2y2 - b1y1 - b1y2;
  float rho2 = (dx * dx + dy * dy) * 0.25f;
  const float FOUR_OVER_PI2 = 4.0f / 9.869604401089358f;
  float dv = atanf(tw / (th + EPSF)) - atanf(pw / (ph + EPSF));
  float v = FOUR_OVER_PI2 * dv * dv;
  float alpha = v / (v - iou + (1.0f + EPSF));
  return iou - (rho2 / c2 + v * alpha);
}

// ---------------------------------------------------------------------------
// Per-target kernel: CIoU box loss, t_obj scatter, class loss
//   cls BCE sum = sum_c softplus(l_c) - l_{target_class}   (pw=1, one-hot)
// ---------------------------------------------------------------------------
__global__ void __launch_bounds__(256)
yolo_target_kernel(const float* __restrict__ outp, const int* __restrict__ img,
                   const int* __restrict__ anc, const int* __restrict__ gj,
                   const int* __restrict__ gi, const float* __restrict__ tgt,
                   const float* __restrict__ A, float* __restrict__ scat,
                   float* __restrict__ box_part, float* __restrict__ cls_part,
                   int nt, int g) {
  __shared__ float shA[8];
  __shared__ float shB[8];

  int t = blockIdx.x * blockDim.x + threadIdx.x;
  bool act = (t < nt);
  int tt = act ? t : 0;

  int vi = img[tt], va = anc[tt], vj = gj[tt], vg = gi[tt];
  long long cell = ((long long)((vi * 3 + va) * g + vj)) * g + vg;
  const float* p = outp + cell * CH;
  __builtin_prefetch(p, 0, 1);            // global_prefetch_b8: whole 340B row
  __builtin_prefetch(p + 64, 0, 1);

  float px  = sigmoidf(p[0]) * 2.0f - 0.5f;
  float py  = sigmoidf(p[1]) * 2.0f - 0.5f;
  float s2w = sigmoidf(p[2]) * 2.0f;
  float s2h = sigmoidf(p[3]) * 2.0f;
  float pw  = s2w * s2w * A[tt * 2 + 0];
  float ph  = s2h * s2h * A[tt * 2 + 1];

  float gf = (float)g;
  float tx = tgt[tt * 6 + 2] * gf - (float)vg;
  float ty = tgt[tt * 6 + 3] * gf - (float)vj;
  float tw = tgt[tt * 6 + 4] * gf;
  float th = tgt[tt * 6 + 5] * gf;

  float iou = ciou_f(px, py, pw, ph, tx, ty, tw, th);

  int tc = (int)tgt[tt * 6 + 1];
  float csum = 0.0f;
  #pragma unroll 4
  for (int c = 0; c < NCLS; ++c) csum += softplusf(p[5 + c]);
  csum -= p[5 + tc];

  if (act) scat[cell] = fmaxf(iou, 0.0f);   // t_obj (IOU_RATIO==1); last-wins like JAX .set

  float boxc = act ? (1.0f - iou) : 0.0f;
  float clsc = act ? csum : 0.0f;

  // EXEC is all-1s here (no early returns) -> WMMA reduction is legal.
  float wb = wmma_wave_sum(boxc);
  float wc = wmma_wave_sum(clsc);

  int lane = threadIdx.x & 31, wid = threadIdx.x >> 5;
  if (lane == 0) { shA[wid] = wb; shB[wid] = wc; }
  __syncthreads();
  if (threadIdx.x == 0) {
    float sb = 0.0f, sc = 0.0f;
    #pragma unroll
    for (int i = 0; i < 8; ++i) { sb += shA[i]; sc += shB[i]; }
    box_part[blockIdx.x] = sb;
    cls_part[blockIdx.x] = sc;
  }
}

// ---------------------------------------------------------------------------
// Objectness kernel over all grid cells of one level.
//   obj BCE sum = sum_cells softplus(x) - t*x ; x = out[cell*85+4]
// The strided channel-4 reads are staged through LDS with the CDNA5 async
// load-to-LDS path (ASYNCcnt), then reduced with WMMA.
// ---------------------------------------------------------------------------
__global__ void __launch_bounds__(256)
yolo_obj_kernel(const float* __restrict__ outp, const float* __restrict__ scat,
                float* __restrict__ obj_part, int cells) {
  __shared__ float stage[256];
  __shared__ float shW[8];

  int c = blockIdx.x * blockDim.x + threadIdx.x;
  bool act = (c < cells);
  int cc = act ? c : (cells - 1);   // clamp: keep async load address in range

  const float* gp = outp + (long long)cc * CH + 4;
  unsigned lds_off = (unsigned)(size_t)(&stage[threadIdx.x]);

  // CDNA5 async global->LDS copy; tracked by ASYNCcnt (per-wave).
  asm volatile("global_load_async_to_lds_b32 %0, %1, off"
               :: "v"(lds_off), "v"(gp)
               : "memory");
  asm volatile("s_wait_asynccnt 0x0" ::: "memory");

  float x  = stage[threadIdx.x];
  float tv = scat[cc];
  float contrib = act ? (softplusf(x) - tv * x) : 0.0f;

  float w = wmma_wave_sum(contrib);
  int lane = threadIdx.x & 31, wid = threadIdx.x >> 5;
  if (lane == 0) shW[wid] = w;
  __syncthreads();
  if (threadIdx.x == 0) {
    float s = 0.0f;
    #pragma unroll
    for (int i = 0; i < 8; ++i) s += shW[i];
    obj_part[blockIdx.x] = s;
  }
}

// ---------------------------------------------------------------------------
// Final combine: deterministic serial reduction of all partials.
// ---------------------------------------------------------------------------
__global__ void yolo_final_kernel(const float* __restrict__ box_part,
                                  const float* __restrict__ cls_part,
                                  const float* __restrict__ obj_part,
                                  float* __restrict__ outp,
                                  int nb0, int nb1, int nb2,
                                  int no0, int no1, int no2,
                                  int nt0, int nt1, int nt2) {
  if (threadIdx.x != 0 || blockIdx.x != 0) return;
  const float bal[3] = {4.0f, 1.0f, 0.4f};
  const int   nts[3] = {nt0, nt1, nt2};
  const int   nbs[3] = {nb0, nb1, nb2};
  const int   nos[3] = {no0, no1, no2};
  const int   gs[3]  = {80, 40, 20};
  float lbox = 0.0f, lobj = 0.0f, lcls = 0.0f;
  for (int i = 0; i < 3; ++i) {
    float sb = 0.0f, sc = 0.0f, so = 0.0f;
    for (int k = 0; k < nbs[i]; ++k) { sb += box_part[i * 32 + k]; sc += cls_part[i * 32 + k]; }
    for (int k = 0; k < nos[i]; ++k) { so += obj_part[i * 1280 + k]; }
    float cellsf = 16.0f * 3.0f * (float)gs[i] * (float)gs[i];
    lbox += sb / (float)nts[i];
    lcls += sc / ((float)nts[i] * (float)NCLS);
    lobj += (so / cellsf) * bal[i];
  }
  lbox *= 0.05f;   // BOX_GAIN * 3/nl
  lobj *= 0.7f;    // OBJ_GAIN * 1 * 1 * 3/nl
  lcls *= 0.3f;    // CLS_GAIN * (NC/80) * 3/nl
  float loss = lbox + lobj + lcls;
  outp[0] = loss * 16.0f;  // * B
  outp[1] = lbox;
  outp[2] = lobj;
  outp[3] = lcls;
  outp[4] = loss;
}

// ---------------------------------------------------------------------------
extern "C" void kernel_launch(void* const* d_in, const int* in_sizes, int n_in,
                              void* d_out, int out_size, void* d_ws, size_t ws_size,
                              hipStream_t stream) {
  (void)n_in; (void)out_size; (void)ws_size;
  float* ws = (float*)d_ws;
  const int gs[3]       = {80, 40, 20};
  const int scat_off[3] = {0, 307200, 384000};  // 16*3*80*80, +16*3*40*40
  const int SCAT_TOTAL  = 403200;

  float* scat     = ws;
  float* box_part = ws + SCAT_TOTAL;          // 3 x 32
  float* cls_part = ws + SCAT_TOTAL + 96;     // 3 x 32
  float* obj_part = ws + SCAT_TOTAL + 192;    // 3 x 1280

  hipMemsetAsync(scat, 0, (size_t)SCAT_TOTAL * sizeof(float), stream);

  int nts[3], nbs[3], nos[3];
  for (int i = 0; i < 3; ++i) {
    int base = i * 7;   // dict order: out, img, anc, gj, gi, tgt, A per level
    const float* outp = (const float*)d_in[base + 0];
    const int*   img  = (const int*)  d_in[base + 1];
    const int*   anc  = (const int*)  d_in[base + 2];
    const int*   gjp  = (const int*)  d_in[base + 3];
    const int*   gip  = (const int*)  d_in[base + 4];
    const float* tgt  = (const float*)d_in[base + 5];
    const float* Ap   = (const float*)d_in[base + 6];
    int nt = in_sizes[base + 1];
    nts[i] = nt;
    nbs[i] = (nt + 255) / 256;
    yolo_target_kernel<<<nbs[i], 256, 0, stream>>>(
        outp, img, anc, gjp, gip, tgt, Ap,
        scat + scat_off[i], box_part + i * 32, cls_part + i * 32, nt, gs[i]);
  }

  for (int i = 0; i < 3; ++i) {
    int base = i * 7;
    const float* outp = (const float*)d_in[base + 0];
    int cells = 16 * 3 * gs[i] * gs[i];
    nos[i] = (cells + 255) / 256;
    yolo_obj_kernel<<<nos[i], 256, 0, stream>>>(
        outp, scat + scat_off[i], obj_part + i * 1280, cells);
  }

  yolo_final_kernel<<<1, 1, 0, stream>>>(
      box_part, cls_part, obj_part, (float*)d_out,
      nbs[0], nbs[1], nbs[2], nos[0], nos[1], nos[2], nts[0], nts[1], nts[2]);
}